// RelationEncoder_60773787238647
// MI455X (gfx1250) — compile-verified
//
#include <hip/hip_runtime.h>
#include <hip/hip_bf16.h>
#include <math.h>

typedef __attribute__((ext_vector_type(2))) float v2f;
typedef __attribute__((ext_vector_type(8))) float v8f;

#define S_NUM   256
#define A_NUM   64
#define FC7     4096
#define JEMB    512
#define CONCAT  (FC7 + 5)
#define EPSN    1e-12f

// ---- workspace layout (element offsets) ----
// ints:   max_id[256] at 0, filt[256] at 256
// floats: scale_row[64] at 512
//         fcwt  (4096*512, K-pair-major, w_fc7 folded) at 1024
//         tail  (5*512, w_lfeat folded)                at 1024+2097152
//         base  (256*512, bias folded)                 at 1024+2097152+2560
#define OFF_MAXID 0
#define OFF_FILT  256
#define OFF_SCALE 512
#define OFF_FCWT  1024
#define OFF_TAIL  (OFF_FCWT + FC7*JEMB)
#define OFF_BASE  (OFF_TAIL + 5*JEMB)

// ------------------------------------------------------------------
// 1) per-sentence argmax (first-max tie-break), dists + max_id outputs
// ------------------------------------------------------------------
__global__ void k_argmax(const float* __restrict__ obj_attn,
                         const float* __restrict__ dist,
                         int* __restrict__ max_id, int* __restrict__ filt,
                         float* __restrict__ out_dists, float* __restrict__ out_maxid) {
    int s = blockIdx.x;
    int t = threadIdx.x;                 // 0..63
    __shared__ float sv[64];
    __shared__ int   si[64];
    sv[t] = obj_attn[s * A_NUM + t];
    si[t] = t;
    __syncthreads();
    for (int off = 32; off > 0; off >>= 1) {
        if (t < off) {
            float v2 = sv[t + off]; int i2 = si[t + off];
            if (v2 > sv[t] || (v2 == sv[t] && i2 < si[t])) { sv[t] = v2; si[t] = i2; }
        }
        __syncthreads();
    }
    int   id = si[0];
    int   f  = (sv[0] == 0.0f) ? 1 : 0;
    if (t == 0) { max_id[s] = id; filt[s] = f; out_maxid[s] = (float)id; }
    // dists[s,a] = filt ? 100 : dist[a, id, 0]
    out_dists[s * A_NUM + t] = f ? 100.0f : dist[t * A_NUM + id];
}

// ------------------------------------------------------------------
// 2) per-row inverse L2 norm of cxt_feats (64 rows x 4096)
// ------------------------------------------------------------------
__global__ void k_rownorm(const float* __restrict__ cxt_feats,
                          float* __restrict__ scale_row) {
    int r = blockIdx.x;                  // 0..63
    int t = threadIdx.x;                 // 0..255
    __shared__ float red[256];
    float acc = 0.0f;
    for (int k = t; k < FC7; k += 256) {
        float x = cxt_feats[r * FC7 + k];
        acc += x * x;
    }
    red[t] = acc; __syncthreads();
    for (int off = 128; off > 0; off >>= 1) {
        if (t < off) red[t] += red[t + off];
        __syncthreads();
    }
    if (t == 0) scale_row[r] = 1.0f / fmaxf(sqrtf(red[0]), EPSN);
}

// ------------------------------------------------------------------
// 3) repack fc_w: first 4096 cols -> K-pair-major (fold w_fc7),
//    last 5 cols -> tail[d][j] (fold w_lfeat)
//    fcwt float2 at pair index (k>>1)*512 + j holds (w*fc_w[j][k], w*fc_w[j][k+1])
// ------------------------------------------------------------------
__global__ void k_split_w(const float* __restrict__ fc_w,
                          const float* __restrict__ w_fc7,
                          const float* __restrict__ w_lfeat,
                          float* __restrict__ fcwt, float* __restrict__ tail) {
    int i = blockIdx.x * 256 + threadIdx.x;
    if (i >= JEMB * CONCAT) return;
    int j = i / CONCAT;
    int k = i - j * CONCAT;
    float v = fc_w[i];
    if (k < FC7) fcwt[(k >> 1) * (2 * JEMB) + 2 * j + (k & 1)] = v * w_fc7[k];
    else         tail[(k - FC7) * JEMB + j]                    = v * w_lfeat[k - FC7];
}

// ------------------------------------------------------------------
// 4) WMMA f32 GEMM: base[256,512] = Aunit[256,4096] @ Wt + fc_b
//    One wave per 16x16 output tile. A row s = cxt_feats[max_id[s]] * scale.
//    A frag (16x4 f32): lanes 0-15 -> K{k,k+1}, lanes 16-31 -> K{k+2,k+3}.
//    B frag (4x16 f32): VGPR0 = K(k + 2*hi), VGPR1 = K(k+1 + 2*hi), N = lane&15.
// ------------------------------------------------------------------
__global__ void k_gemm(const float* __restrict__ cxt_feats,
                       const float* __restrict__ fcwt,
                       const float* __restrict__ fc_b,
                       const int* __restrict__ max_id,
                       const int* __restrict__ filt,
                       const float* __restrict__ scale_row,
                       float* __restrict__ base) {
    int s0   = blockIdx.x * 16;
    int j0   = blockIdx.y * 16;
    int lane = threadIdx.x;              // 0..31
    int m    = lane & 15;
    int hi   = lane >> 4;

    int   row = s0 + m;
    int   id  = max_id[row];
    float sc  = filt[row] ? 0.0f : scale_row[id];

    const float*  arow = cxt_feats + (size_t)id * FC7 + 2 * hi;
    const float2* bcol = ((const float2*)fcwt) + (j0 + m);

    v8f c = {0.f, 0.f, 0.f, 0.f, 0.f, 0.f, 0.f, 0.f};
#pragma unroll 4
    for (int k = 0; k < FC7; k += 4) {
        float2 av = *(const float2*)(arow + k);
        v2f a = { av.x * sc, av.y * sc };
        float2 bv = bcol[(size_t)((k + 2 * hi) >> 1) * JEMB];
        v2f b = { bv.x, bv.y };
        c = __builtin_amdgcn_wmma_f32_16x16x4_f32(
                /*neg_a=*/false, a, /*neg_b=*/false, b,
                /*c_mod=*/(short)0, c, /*reuse_a=*/false, /*reuse_b=*/false);
    }

    int   col  = j0 + m;
    float bias = fc_b[col];
    // C/D layout: VGPR i -> row M=i (lanes 0-15) / M=i+8 (lanes 16-31), N=lane&15
#pragma unroll
    for (int i = 0; i < 8; ++i) {
        int r = s0 + i + 8 * hi;
        base[r * JEMB + col] = c[i] + bias;
    }
}

// ------------------------------------------------------------------
// 5) fuse[s,a,:] = base[s,:] + normalize(cxt_lfeats[a, max_id[s], :]) @ tail
// ------------------------------------------------------------------
__global__ void k_fuse(const float* __restrict__ cxt_lfeats,
                       const float* __restrict__ base,
                       const float* __restrict__ tail,
                       const int* __restrict__ max_id,
                       const int* __restrict__ filt,
                       float* __restrict__ out_fuse) {
    int bid = blockIdx.x;                // s*64 + a
    int s = bid >> 6;
    int a = bid & 63;
    int t = threadIdx.x;                 // 0..255

    int id = max_id[s];
    int f  = filt[s];

    float lf[5];
    const float* lp = cxt_lfeats + (size_t)(a * A_NUM + id) * 5;
    float nsq = 0.0f;
#pragma unroll
    for (int d = 0; d < 5; ++d) { lf[d] = lp[d]; nsq += lf[d] * lf[d]; }
    float inv = f ? 0.0f : (1.0f / fmaxf(sqrtf(nsq), EPSN));
#pragma unroll
    for (int d = 0; d < 5; ++d) lf[d] *= inv;

    const float* bs = base + s * JEMB;
    float*       op = out_fuse + (size_t)bid * JEMB;
    for (int j = t; j < JEMB; j += 256) {
        float acc = bs[j];
#pragma unroll
        for (int d = 0; d < 5; ++d) acc += lf[d] * tail[d * JEMB + j];
        op[j] = acc;
    }
}

// ------------------------------------------------------------------
extern "C" void kernel_launch(void* const* d_in, const int* in_sizes, int n_in,
                              void* d_out, int out_size, void* d_ws, size_t ws_size,
                              hipStream_t stream) {
    const float* cxt_feats  = (const float*)d_in[0];   // [64,4096]
    const float* cxt_lfeats = (const float*)d_in[1];   // [64,64,5]
    const float* obj_attn   = (const float*)d_in[2];   // [256,64]
    /* d_in[3] = wo_obj_idx: unused by the reference */
    const float* dist       = (const float*)d_in[4];   // [64,64,1]
    const float* w_fc7      = (const float*)d_in[5];   // [1,4096]
    const float* w_lfeat    = (const float*)d_in[6];   // [1,5]
    const float* fc_w       = (const float*)d_in[7];   // [512,4101]
    const float* fc_b       = (const float*)d_in[8];   // [512]

    float* ws_f    = (float*)d_ws;
    int*   ws_i    = (int*)d_ws;
    int*   max_id  = ws_i + OFF_MAXID;
    int*   filt    = ws_i + OFF_FILT;
    float* scale   = ws_f + OFF_SCALE;
    float* fcwt    = ws_f + OFF_FCWT;
    float* tail    = ws_f + OFF_TAIL;
    float* base    = ws_f + OFF_BASE;

    float* out_fuse  = (float*)d_out;                       // 256*64*512
    float* out_dists = out_fuse + (size_t)S_NUM * A_NUM * JEMB;  // 256*64
    float* out_maxid = out_dists + S_NUM * A_NUM;                // 256

    k_argmax<<<S_NUM, A_NUM, 0, stream>>>(obj_attn, dist, max_id, filt,
                                          out_dists, out_maxid);
    k_rownorm<<<A_NUM, 256, 0, stream>>>(cxt_feats, scale);
    {
        int total = JEMB * CONCAT;
        k_split_w<<<(total + 255) / 256, 256, 0, stream>>>(fc_w, w_fc7, w_lfeat,
                                                           fcwt, tail);
    }
    k_gemm<<<dim3(S_NUM / 16, JEMB / 16), 32, 0, stream>>>(cxt_feats, fcwt, fc_b,
                                                           max_id, filt, scale, base);
    k_fuse<<<S_NUM * A_NUM, 256, 0, stream>>>(cxt_lfeats, base, tail,
                                              max_id, filt, out_fuse);
}